// HSTGNN_7172595384477
// MI455X (gfx1250) — compile-verified
//
#include <hip/hip_runtime.h>

typedef __attribute__((ext_vector_type(2))) float v2f;
typedef __attribute__((ext_vector_type(8))) float v8f;

#define NB 16
#define NS 800
#define NT 224
#define NN 1024   // NS + NT
#define DD 32

// ---------------------------------------------------------------------------
// Kernel 0: pack the 8 weight vectors per node-type into a zero-padded
// [2][16][32] B-matrix buffer. Column layout chosen so phase-2 loads pairs:
//   n=0: row-same  rel   n=1: row-same  mask
//   n=2: row-cross rel   n=3: row-cross mask
//   n=4: col-same  rel   n=5: col-same  mask
//   n=6: col-cross rel   n=7: col-cross mask
// spatial:  same=ss(k0), row-cross=st(k2) first half, col-cross=ts(k3) 2nd half
// temporal: same=tt(k1), row-cross=ts(k3) first half, col-cross=st(k2) 2nd half
// ---------------------------------------------------------------------------
__global__ void hstgnn_pack_weights(const float* __restrict__ relW,
                                    const float* __restrict__ maskW,
                                    float* __restrict__ wpack) {
    int i = blockIdx.x * blockDim.x + threadIdx.x;   // 0 .. 1023
    if (i >= 2 * 16 * 32) return;
    int d    = i & 31;
    int n    = (i >> 5) & 15;
    int type = i >> 9;
    float v = 0.0f;
    if (n < 8) {
        const int krow[2][8] = { {0,0,2,2,0,0,3,3},
                                 {1,1,3,3,1,1,2,2} };
        int k    = krow[type][n];
        int half = (n >> 2) & 1;                     // 0: W[:D], 1: W[D:]
        const float* src = (n & 1) ? maskW : relW;   // rel_W / mask_W are [4][64]
        v = src[k * 64 + half * 32 + d];
    }
    wpack[i] = v;
}

// ---------------------------------------------------------------------------
// Kernel 1: per-node score table via V_WMMA_F32_16X16X4_F32.
// One wave per 16-node tile: D(16x16) = A(16x32 nodes) x B(32x16 weights),
// accumulated over 8 WMMAs of K=4. Scores stored as [B][1024][16] (cols 8..15
// are the zero pad).
// ---------------------------------------------------------------------------
__global__ void hstgnn_score_wmma(const float* __restrict__ sp,
                                  const float* __restrict__ tp,
                                  const float* __restrict__ wpack,
                                  float* __restrict__ scores) {
    int id   = blockIdx.x;            // 0 .. NB*64 - 1
    int b    = id >> 6;
    int t    = id & 63;               // tile within combined 1024-node axis
    int lane = threadIdx.x;           // 0..31 (wave32)
    int r    = lane & 15;
    int hi   = lane >> 4;             // 0: K=(0,1), 1: K=(2,3) per WMMA step

    const float* src;
    int type;
    if (t < 50) { type = 0; src = sp + ((size_t)b * NS + (size_t)t * 16) * DD; }
    else        { type = 1; src = tp + ((size_t)b * NT + (size_t)(t - 50) * 16) * DD; }
    const float* w = wpack + type * (16 * 32);

    v8f acc = {0.f,0.f,0.f,0.f,0.f,0.f,0.f,0.f};
#pragma unroll
    for (int k = 0; k < 8; ++k) {
        int kb = k * 4 + hi * 2;      // even -> 8B-aligned float2 loads
        v2f a  = *(const v2f*)(src + r * DD + kb);  // A[M=r][K=kb..kb+1]
        v2f bb = *(const v2f*)(w   + r * DD + kb);  // B[K][N=r] = Wpack[n=r][K]
        acc = __builtin_amdgcn_wmma_f32_16x16x4_f32(
                  false, a, false, bb, (short)0, acc, false, false);
    }

    float* dst = scores + ((size_t)b * NN + (size_t)t * 16) * 16;
#pragma unroll
    for (int v = 0; v < 8; ++v) {
        int row = v + hi * 8;         // C/D layout: VGPR v -> M = v (+8 hi half)
        dst[row * 16 + r] = acc[v];
    }
}

// ---------------------------------------------------------------------------
// Kernel 2: emit 64 MiB output (store-bandwidth bound).
// One block per (b, i) row; each thread writes a float4 of columns.
// ---------------------------------------------------------------------------
__global__ void hstgnn_emit(const float* __restrict__ scores,
                            const float* __restrict__ relb,
                            const float* __restrict__ maskb,
                            float* __restrict__ out) {
    int bi = blockIdx.x;              // 0 .. NB*NN - 1
    int b  = bi >> 10;
    int i  = bi & 1023;
    int j0 = threadIdx.x * 4;         // 800 % 4 == 0: quad never straddles S/T
    bool iT = (i  >= NS);
    bool jT = (j0 >= NS);

    int rowslot = (iT == jT) ? 0 : 2;
    int colslot = (iT == jT) ? 4 : 6;
    // bias row index: ss=0 tt=1 st=2 ts=3
    int k = jT ? (iT ? 1 : 2) : (iT ? 3 : 0);

    const float* srow = scores + ((size_t)b * NN + i) * 16;
    float2 rv = *(const float2*)(srow + rowslot);   // (rel_row, mask_row)
    float  rb = relb[k];
    float  mb = maskb[k];

    const float* scol = scores + (size_t)b * NN * 16;
    float res[4];
#pragma unroll
    for (int u = 0; u < 4; ++u) {
        int j = j0 + u;
        float2 cv = *(const float2*)(scol + (size_t)j * 16 + colslot);
        float adj  = rv.x + cv.x + rb;
        float mask = fmaxf(rv.y + cv.y + mb, 0.0f);
        res[u] = tanhf(adj * mask);
    }
    *(float4*)(out + (size_t)bi * NN + j0) =
        make_float4(res[0], res[1], res[2], res[3]);
}

// ---------------------------------------------------------------------------
extern "C" void kernel_launch(void* const* d_in, const int* in_sizes, int n_in,
                              void* d_out, int out_size, void* d_ws, size_t ws_size,
                              hipStream_t stream) {
    const float* spatial  = (const float*)d_in[0];  // [16,800,32]
    const float* temporal = (const float*)d_in[1];  // [16,224,32]
    const float* rel_W    = (const float*)d_in[2];  // [4,64]
    const float* rel_b    = (const float*)d_in[3];  // [4]
    const float* mask_W   = (const float*)d_in[4];  // [4,64]
    const float* mask_b   = (const float*)d_in[5];  // [4]
    float*       out      = (float*)d_out;          // [16,1024,1024]

    float* wpack  = (float*)d_ws;                   // 2*16*32 floats = 4 KiB
    float* scores = wpack + 2 * 16 * 32;            // 16*1024*16 floats = 1 MiB

    hipLaunchKernelGGL(hstgnn_pack_weights, dim3(4), dim3(256), 0, stream,
                       rel_W, mask_W, wpack);
    hipLaunchKernelGGL(hstgnn_score_wmma, dim3(NB * 64), dim3(32), 0, stream,
                       spatial, temporal, wpack, scores);
    hipLaunchKernelGGL(hstgnn_emit, dim3(NB * NN), dim3(256), 0, stream,
                       scores, rel_b, mask_b, out);
}